// Network_IJCAI_54820962566210
// MI455X (gfx1250) — compile-verified
//
#include <hip/hip_runtime.h>
#include <hip/hip_bf16.h>
#include <stdint.h>

#define NB     8
#define NCLS   80
#define NANCH  13343
#define MSORT  16384
#define TOPK   1000
#define SCORE_THR 0.05f
#define IOU_THR   0.6f

struct LevelsPtrs {
    const float* cls[5];
    const float* cnt[5];
    const float* reg[5];
};

__device__ __forceinline__ float sigmoidf(float x) {
    return 1.0f / (1.0f + __expf(-x));
}

// ---- CDNA5 async global->LDS copy (ASYNCcnt path) -------------------------
// LDS byte offset = low 32 bits of the flat address.
__device__ __forceinline__ void async_b32_to_lds(const void* g, const void* l) {
    unsigned           lds = (unsigned)(unsigned long long)l;
    unsigned long long ga  = (unsigned long long)g;
    asm volatile("global_load_async_to_lds_b32 %0, %1, off"
                 :: "v"(lds), "v"(ga) : "memory");
}
__device__ __forceinline__ void wait_asynccnt0() {
    asm volatile("s_wait_asynccnt 0" ::: "memory");
}

// ---------------------------------------------------------------------------
// Kernel 1: per-anchor decode. score = sqrt(sigmoid(max_c logit)*sigmoid(cnt)),
// class = argmax+1 (first max), box = coords -/+ reg.  NCHW layout handled by
// per-class strided reads (coalesced across hw within each class plane).
// ---------------------------------------------------------------------------
__global__ __launch_bounds__(256)
void decode_kernel(LevelsPtrs P, float* __restrict__ scores_all,
                   int* __restrict__ cls_all, float* __restrict__ box_all)
{
    int t = blockIdx.x * blockDim.x + threadIdx.x;
    if (t >= NB * NANCH) return;
    int b = t / NANCH;
    int a = t - b * NANCH;

    int lvl, start, W, stride, HW;
    if      (a < 10000) { lvl = 0; start = 0;     W = 100; stride = 8;   HW = 10000; }
    else if (a < 12500) { lvl = 1; start = 10000; W = 50;  stride = 16;  HW = 2500;  }
    else if (a < 13125) { lvl = 2; start = 12500; W = 25;  stride = 32;  HW = 625;   }
    else if (a < 13294) { lvl = 3; start = 13125; W = 13;  stride = 64;  HW = 169;   }
    else                { lvl = 4; start = 13294; W = 7;   stride = 128; HW = 49;    }

    int la = a - start;
    int h  = la / W;
    int w  = la - h * W;
    float cx = (float)(w * stride + (stride >> 1));
    float cy = (float)(h * stride + (stride >> 1));

    const float* cp = P.cls[lvl] + (size_t)b * NCLS * HW + la;
    float best = cp[0];
    int   bi   = 0;
#pragma unroll 8
    for (int c = 1; c < NCLS; ++c) {
        float v = cp[(size_t)c * HW];
        if (v > best) { best = v; bi = c; }
    }
    float sc    = sigmoidf(best);
    float ct    = sigmoidf(P.cnt[lvl][(size_t)b * HW + la]);
    float score = sqrtf(sc * ct);

    const float* rp = P.reg[lvl] + (size_t)b * 4 * HW + la;
    float r0 = rp[0];
    float r1 = rp[(size_t)HW];
    float r2 = rp[(size_t)2 * HW];
    float r3 = rp[(size_t)3 * HW];

    scores_all[t] = score;
    cls_all[t]    = bi + 1;
    float* bo = box_all + (size_t)t * 4;
    bo[0] = cx - r0; bo[1] = cy - r1; bo[2] = cx + r2; bo[3] = cy + r3;
}

// ---------------------------------------------------------------------------
// Kernel 2: per-image top-1000 via full in-LDS bitonic sort of 16384 packed
// u64 keys (128 KB LDS -- CDNA5 WGP has 320 KB). key = score_bits<<32 | ~idx:
// scores >= 0 so raw f32 bits are order-preserving; ~idx gives lowest-index
// tie-break, matching lax.top_k. Descending sort; first 1000 are the top-k.
// ---------------------------------------------------------------------------
__global__ __launch_bounds__(1024)
void topk_kernel(const float* __restrict__ scores_all,
                 const int*   __restrict__ cls_all,
                 const float* __restrict__ box_all,
                 float* __restrict__ tk_score, int* __restrict__ tk_cls,
                 float* __restrict__ tk_x1, float* __restrict__ tk_y1,
                 float* __restrict__ tk_x2, float* __restrict__ tk_y2)
{
    __shared__ unsigned long long keys[MSORT];
    const int b   = blockIdx.x;
    const int tid = threadIdx.x;
    const float* sp = scores_all + (size_t)b * NANCH;

    if (tid < NANCH) __builtin_prefetch(sp + tid, 0, 3);   // global_prefetch_b8

    for (int i = tid; i < MSORT; i += 1024) {
        unsigned long long key;
        if (i < NANCH) {
            unsigned sb = __float_as_uint(sp[i]);
            key = ((unsigned long long)sb << 32) | (unsigned)(~(unsigned)i);
        } else {
            key = (unsigned long long)(unsigned)(~0x00FFFFFFu);  // pad: score 0, huge idx
        }
        keys[i] = key;
    }

    for (unsigned k = 2; k <= (unsigned)MSORT; k <<= 1) {
        for (unsigned j = k >> 1; j > 0; j >>= 1) {
            __syncthreads();
            for (unsigned i = tid; i < (unsigned)MSORT; i += 1024) {
                unsigned ixj = i ^ j;
                if (ixj > i) {
                    unsigned long long va = keys[i];
                    unsigned long long vb = keys[ixj];
                    bool desc = ((i & k) == 0);
                    if (desc ? (va < vb) : (va > vb)) {
                        keys[i] = vb; keys[ixj] = va;
                    }
                }
            }
        }
    }
    __syncthreads();

    if (tid < TOPK) {
        unsigned long long key = keys[tid];
        float    score = __uint_as_float((unsigned)(key >> 32));
        unsigned idx   = ~(unsigned)key;
        size_t g = (size_t)b * NANCH + idx;
        size_t o = (size_t)b * TOPK + tid;
        tk_score[o] = score;
        tk_cls[o]   = cls_all[g];
        const float* bo = box_all + g * 4;
        tk_x1[o] = bo[0]; tk_y1[o] = bo[1]; tk_x2[o] = bo[2]; tk_y2[o] = bo[3];
    }
}

// ---------------------------------------------------------------------------
// Kernel 3: greedy class-aware NMS, one workgroup per image. Candidates are
// staged into LDS with CDNA5 async global->LDS loads + s_wait_asynccnt.
// Exactly mirrors the reference: valid = score>=thr; max_coord over
// where(valid, boxes, 0); offset = class*(max_coord+1); serial suppression.
// ---------------------------------------------------------------------------
__global__ __launch_bounds__(1024)
void nms_kernel(const float* __restrict__ tk_score, const int* __restrict__ tk_cls,
                const float* __restrict__ tk_x1, const float* __restrict__ tk_y1,
                const float* __restrict__ tk_x2, const float* __restrict__ tk_y2,
                float* __restrict__ out)
{
    __shared__ float sx1[1024], sy1[1024], sx2[1024], sy2[1024];
    __shared__ float ssc[1024], sar[1024], red[1024];
    __shared__ int   scl[1024], skeep[1024];

    const int b   = blockIdx.x;
    const int tid = threadIdx.x;
    const size_t o = (size_t)b * TOPK + tid;

    if (tid < TOPK) {
        async_b32_to_lds(tk_x1 + o,    &sx1[tid]);
        async_b32_to_lds(tk_y1 + o,    &sy1[tid]);
        async_b32_to_lds(tk_x2 + o,    &sx2[tid]);
        async_b32_to_lds(tk_y2 + o,    &sy2[tid]);
        async_b32_to_lds(tk_score + o, &ssc[tid]);
        async_b32_to_lds(tk_cls + o,   &scl[tid]);
    }
    wait_asynccnt0();
    __syncthreads();

    float ox1 = 0.f, oy1 = 0.f, ox2 = 0.f, oy2 = 0.f, score = 0.f;
    int   cls = 0;
    bool  valid = false;
    float contrib = -INFINITY;
    if (tid < TOPK) {
        ox1 = sx1[tid]; oy1 = sy1[tid]; ox2 = sx2[tid]; oy2 = sy2[tid];
        score = ssc[tid]; cls = scl[tid];
        valid = (score >= SCORE_THR);
        float m4 = fmaxf(fmaxf(ox1, oy1), fmaxf(ox2, oy2));
        contrib = valid ? m4 : 0.0f;   // where(valid, boxes, 0) row contribution
    }
    red[tid] = contrib;
    for (int s = 512; s > 0; s >>= 1) {
        __syncthreads();
        if (tid < s) red[tid] = fmaxf(red[tid], red[tid + s]);
    }
    __syncthreads();
    float mc = red[0];
    __syncthreads();

    if (tid < TOPK) {
        float off = (float)cls * (mc + 1.0f);
        float x1 = ox1 + off, y1 = oy1 + off, x2 = ox2 + off, y2 = oy2 + off;
        sx1[tid] = x1; sy1[tid] = y1; sx2[tid] = x2; sy2[tid] = y2;
        sar[tid] = (x2 - x1 + 1.0f) * (y2 - y1 + 1.0f);
        skeep[tid] = valid ? 1 : 0;
    } else {
        skeep[tid] = 0;
    }
    __syncthreads();

    for (int i = 0; i < TOPK; ++i) {
        if (skeep[i]) {                       // uniform read; never written this step
            int j = tid;
            if (j < TOPK && j > i && skeep[j]) {
                float xmin = fmaxf(sx1[j], sx1[i]);
                float ymin = fmaxf(sy1[j], sy1[i]);
                float xmax = fminf(sx2[j], sx2[i]);
                float ymax = fminf(sy2[j], sy2[i]);
                float inter = fmaxf(xmax - xmin, 0.0f) * fmaxf(ymax - ymin, 0.0f);
                float iou   = inter / (sar[i] + sar[j] - inter);
                if (iou > IOU_THR) skeep[j] = 0;
            }
        }
        __syncthreads();
    }

    if (tid < TOPK) {
        bool k = (skeep[tid] != 0);
        float* out_s = out;                       // (8,1000) scores
        float* out_c = out + NB * TOPK;           // (8,1000) classes (as f32)
        float* out_b = out + 2 * NB * TOPK;       // (8,1000,4) boxes
        out_s[o] = k ? score : 0.0f;
        out_c[o] = k ? (float)cls : 0.0f;
        float* bp = out_b + o * 4;
        bp[0] = k ? ox1 : 0.0f;
        bp[1] = k ? oy1 : 0.0f;
        bp[2] = k ? ox2 : 0.0f;
        bp[3] = k ? oy2 : 0.0f;
    }
}

extern "C" void kernel_launch(void* const* d_in, const int* in_sizes, int n_in,
                              void* d_out, int out_size, void* d_ws, size_t ws_size,
                              hipStream_t stream) {
    (void)in_sizes; (void)n_in; (void)out_size; (void)ws_size;

    LevelsPtrs P;
    for (int i = 0; i < 5; ++i) {
        P.cls[i] = (const float*)d_in[i];
        P.cnt[i] = (const float*)d_in[5 + i];
        P.reg[i] = (const float*)d_in[10 + i];
    }

    char* ws = (char*)d_ws;
    const size_t BN = (size_t)NB * NANCH;
    float* scores_all = (float*)ws; ws += BN * 4;
    int*   cls_all    = (int*)ws;   ws += BN * 4;
    float* box_all    = (float*)ws; ws += BN * 16;
    float* tk_score   = (float*)ws; ws += (size_t)NB * TOPK * 4;
    int*   tk_cls     = (int*)ws;   ws += (size_t)NB * TOPK * 4;
    float* tk_x1      = (float*)ws; ws += (size_t)NB * TOPK * 4;
    float* tk_y1      = (float*)ws; ws += (size_t)NB * TOPK * 4;
    float* tk_x2      = (float*)ws; ws += (size_t)NB * TOPK * 4;
    float* tk_y2      = (float*)ws; ws += (size_t)NB * TOPK * 4;

    const int total = NB * NANCH;
    decode_kernel<<<(total + 255) / 256, 256, 0, stream>>>(P, scores_all, cls_all, box_all);
    topk_kernel<<<NB, 1024, 0, stream>>>(scores_all, cls_all, box_all,
                                         tk_score, tk_cls, tk_x1, tk_y1, tk_x2, tk_y2);
    nms_kernel<<<NB, 1024, 0, stream>>>(tk_score, tk_cls, tk_x1, tk_y1, tk_x2, tk_y2,
                                        (float*)d_out);
}